// DMFM_Wei2022_22024592293912
// MI455X (gfx1250) — compile-verified
//
#include <hip/hip_runtime.h>
#include <math.h>

#define NN   50000
#define FF   128
#define HID  64
#define HEADS 2
#define EE   800000
#define NEG  0.2f
#define BN_EPS 1e-5f
#define NBN  256   // partial-sum blocks for batchnorm

typedef __attribute__((ext_vector_type(2))) float v2f;
typedef __attribute__((ext_vector_type(8))) float v8f;

// ---------------- order-preserving float<->uint for atomic max ----------------
__device__ __forceinline__ unsigned fordered(float f) {
    unsigned u = __float_as_uint(f);
    return (u & 0x80000000u) ? ~u : (u | 0x80000000u);
}
__device__ __forceinline__ float funordered(unsigned u) {
    u = (u & 0x80000000u) ? (u & 0x7fffffffu) : ~u;
    return __uint_as_float(u);
}

// ---------------- BatchNorm: deterministic two-stage reduction ----------------
__global__ void bn_partial(const float* __restrict__ x,
                           float* __restrict__ psum, float* __restrict__ psq) {
    int f = threadIdx.x;            // 0..127 (feature) -> coalesced
    int b = blockIdx.x;             // 0..NBN-1
    float s = 0.f, q = 0.f;
    for (int r = b; r < NN; r += NBN) {
        float v = x[r * FF + f];
        s += v; q += v * v;
    }
    psum[b * FF + f] = s;
    psq [b * FF + f] = q;
}

__global__ void bn_finalize(const float* __restrict__ psum, const float* __restrict__ psq,
                            const float* __restrict__ gamma, const float* __restrict__ beta,
                            float* __restrict__ scale, float* __restrict__ shift) {
    int f = threadIdx.x;            // 128 threads
    float s = 0.f, q = 0.f;
    for (int b = 0; b < NBN; ++b) { s += psum[b * FF + f]; q += psq[b * FF + f]; }
    float mean = s / (float)NN;
    float var  = q / (float)NN - mean * mean;
    float rstd = rsqrtf(var + BN_EPS);
    float sc = gamma[f] * rstd;
    scale[f] = sc;
    shift[f] = beta[f] - mean * sc;
}

// ---------------- WMMA f32 GEMM: C = act(Agather[M,K] @ W[K,Nc] + bias) --------
// One wave computes a 16x64 output stripe: 4x v_wmma_f32_16x16x4_f32 per K-step,
// sharing a single A fragment across the 4 column tiles (A streamed once).
// ACT:   0=none, 1=relu, 2=leaky_relu(0.2)
// AMODE: 0=plain A0[M,K]; 1=batchnorm fused (A0*scale+shift); 2=concat3(A0|A1|A2)
template <int ACT, int AMODE>
__global__ void wmma_gemm(const float* __restrict__ A0, const float* __restrict__ A1,
                          const float* __restrict__ A2,
                          const float* __restrict__ bnsc, const float* __restrict__ bnsh,
                          const float* __restrict__ W, const float* __restrict__ bias,
                          float* __restrict__ C, int K, int Nc) {
    const int wave = threadIdx.x >> 5;                  // 4 waves / block, 4 row tiles
    const int lane = threadIdx.x & 31;
    const int m0 = (blockIdx.x * 4 + wave) * 16;
    if (m0 >= NN) return;                               // wave-uniform exit
    const int n0 = blockIdx.y * 64;
    const int m  = lane & 15;
    const int n  = lane & 15;
    const int kh = (lane >> 4) << 1;                    // 0 or 2 (K sub-pair)

    v8f acc[4] = {};
    for (int k = 0; k < K; k += 4) {
        v2f a;
        if (AMODE == 2) {
            // k+kh and k+kh+1 stay inside one 64-aligned block (k%4==0, kh<=2)
            const int kk = k + kh;
            const float* __restrict__ src = (kk < 64) ? A0 : (kk < 128) ? A1 : A2;
            a = *(const v2f*)(src + (size_t)(m0 + m) * HID + (kk & 63));
        } else {
            a = *(const v2f*)(A0 + (size_t)(m0 + m) * K + k + kh);
            if (AMODE == 1) {
                v2f sc = *(const v2f*)(bnsc + k + kh);
                v2f sh = *(const v2f*)(bnsh + k + kh);
                a = a * sc + sh;
            }
        }
#pragma unroll
        for (int t = 0; t < 4; ++t) {                   // 4 independent col tiles
            v2f b;
            b.x = W[(size_t)(k + kh) * Nc + n0 + t * 16 + n];
            b.y = W[(size_t)(k + kh + 1) * Nc + n0 + t * 16 + n];
            acc[t] = __builtin_amdgcn_wmma_f32_16x16x4_f32(
                /*neg_a=*/false, a, /*neg_b=*/false, b,
                /*c_mod=*/(short)0, acc[t], /*reuse_a=*/false, /*reuse_b=*/false);
        }
    }
    const int rbase = m0 + ((lane >> 4) << 3);          // +8 rows for upper lane half
#pragma unroll
    for (int t = 0; t < 4; ++t) {
        const int col = n0 + t * 16 + n;
        const float bcol = bias ? bias[col] : 0.f;
#pragma unroll
        for (int j = 0; j < 8; ++j) {
            float v = acc[t][j] + bcol;
            if (ACT == 1) v = fmaxf(v, 0.f);
            if (ACT == 2) v = (v > 0.f) ? v : NEG * v;
            C[(size_t)(rbase + j) * Nc + col] = v;
        }
    }
}

// ---------------- GAT pieces ----------------
__global__ void att_dots(const float* __restrict__ xl,
                         const float* __restrict__ att_src, const float* __restrict__ att_dst,
                         float* __restrict__ a_src, float* __restrict__ a_dst) {
    int i = blockIdx.x * blockDim.x + threadIdx.x;      // n*HEADS + h
    if (i >= NN * HEADS) return;
    int n = i >> 1, h = i & 1;
    const float* row = xl + (size_t)n * (HEADS * HID) + h * HID;
    const float* ws  = att_src + h * HID;
    const float* wd  = att_dst + h * HID;
    float ss = 0.f, sd = 0.f;
#pragma unroll 4
    for (int c = 0; c < HID; ++c) { float v = row[c]; ss += v * ws[c]; sd += v * wd[c]; }
    a_src[i] = ss; a_dst[i] = sd;
}

__global__ void gat_init(unsigned* __restrict__ emax_bits, float* __restrict__ denom,
                         float* __restrict__ accum) {
    int i = blockIdx.x * blockDim.x + threadIdx.x;
    if (i < NN * HEADS) { emax_bits[i] = fordered(-INFINITY); denom[i] = 0.f; }
    if (i < NN * HEADS * HID) accum[i] = 0.f;
}

__global__ void edge_phase1(const int* __restrict__ src_a, const int* __restrict__ dst_a,
                            const float* __restrict__ a_src, const float* __restrict__ a_dst,
                            float* __restrict__ ebuf, unsigned* __restrict__ emax_bits) {
    int e = blockIdx.x * blockDim.x + threadIdx.x;
    if (e >= EE + NN) return;
    int s, d;
    if (e < EE) { s = src_a[e]; d = dst_a[e]; } else { s = d = e - EE; }  // self loops
    v2f as = *(const v2f*)(a_src + (size_t)s * HEADS);
    v2f ad = *(const v2f*)(a_dst + (size_t)d * HEADS);
    v2f v;
    v.x = as.x + ad.x; v.x = (v.x > 0.f) ? v.x : NEG * v.x;
    v.y = as.y + ad.y; v.y = (v.y > 0.f) ? v.y : NEG * v.y;
    *(v2f*)(ebuf + (size_t)e * HEADS) = v;
    atomicMax(&emax_bits[d * HEADS + 0], fordered(v.x));
    atomicMax(&emax_bits[d * HEADS + 1], fordered(v.y));
}

__global__ void edge_phase2(const int* __restrict__ dst_a, float* __restrict__ ebuf,
                            const unsigned* __restrict__ emax_bits, float* __restrict__ denom) {
    int e = blockIdx.x * blockDim.x + threadIdx.x;
    if (e >= EE + NN) return;
    int d = (e < EE) ? dst_a[e] : e - EE;
    v2f v = *(const v2f*)(ebuf + (size_t)e * HEADS);
    v.x = expf(v.x - funordered(emax_bits[d * HEADS + 0]));
    v.y = expf(v.y - funordered(emax_bits[d * HEADS + 1]));
    *(v2f*)(ebuf + (size_t)e * HEADS) = v;
    atomicAdd(&denom[d * HEADS + 0], v.x);
    atomicAdd(&denom[d * HEADS + 1], v.y);
}

// one wave32 per edge: lane = (head, 4-channel chunk), float4 gather + 4 atomics
__global__ void edge_phase3(const int* __restrict__ src_a, const int* __restrict__ dst_a,
                            const float* __restrict__ ebuf, const float* __restrict__ denom,
                            const float* __restrict__ xl, float* __restrict__ accum) {
    int e = blockIdx.x * 4 + (threadIdx.x >> 5);
    if (e >= EE + NN) return;
    int lane = threadIdx.x & 31;
    int s, d;
    if (e < EE) { s = src_a[e]; d = dst_a[e]; } else { s = d = e - EE; }
    int h  = lane >> 4;
    int cb = (lane & 15) << 2;                           // 4-channel chunk
    float alpha = ebuf[(size_t)e * HEADS + h] / denom[d * HEADS + h];
    const float4 v = *(const float4*)(xl + (size_t)s * (HEADS * HID) + h * HID + cb);
    float* dst = accum + (size_t)d * (HEADS * HID) + h * HID + cb;
    atomicAdd(dst + 0, v.x * alpha);
    atomicAdd(dst + 1, v.y * alpha);
    atomicAdd(dst + 2, v.z * alpha);
    atomicAdd(dst + 3, v.w * alpha);
}

// H = elu(mean_heads(accum) + bias); Cout = Cin - H
__global__ void gat_finalize(const float* __restrict__ accum, const float* __restrict__ bias,
                             const float* __restrict__ Cin, float* __restrict__ Cout) {
    int i = blockIdx.x * blockDim.x + threadIdx.x;      // N*HID
    if (i >= NN * HID) return;
    int n = i >> 6, c = i & 63;
    float v = 0.5f * (accum[(size_t)n * 128 + c] + accum[(size_t)n * 128 + 64 + c]) + bias[c];
    v = (v > 0.f) ? v : (expf(v) - 1.f);                // elu
    Cout[i] = Cin[i] - v;
}

// ---------------- decoder / attention tail ----------------
__global__ void dec2(const float* __restrict__ h2, const float* __restrict__ w,
                     const float* __restrict__ b, float* __restrict__ out) {
    int r = blockIdx.x * blockDim.x + threadIdx.x;
    if (r >= NN) return;
    float s = b[0];
#pragma unroll 4
    for (int i = 0; i < HID; ++i) s += h2[(size_t)r * HID + i] * w[i];
    out[r] = s;
}

__global__ void softmax_rows(float* __restrict__ U) {   // one 128-thr block per row
    __shared__ float red[FF];
    int r = blockIdx.x, t = threadIdx.x;
    float v = U[(size_t)r * FF + t];
    red[t] = v; __syncthreads();
    for (int s = FF / 2; s > 0; s >>= 1) { if (t < s) red[t] = fmaxf(red[t], red[t + s]); __syncthreads(); }
    float m = red[0]; __syncthreads();
    float ex = expf(v - m);
    red[t] = ex; __syncthreads();
    for (int s = FF / 2; s > 0; s >>= 1) { if (t < s) red[t] += red[t + s]; __syncthreads(); }
    U[(size_t)r * FF + t] = ex / red[0];
}

// =======================================================================
extern "C" void kernel_launch(void* const* d_in, const int* in_sizes, int n_in,
                              void* d_out, int out_size, void* d_ws, size_t ws_size,
                              hipStream_t stream) {
    const float* x      = (const float*)d_in[0];
    const int*   ind_ei = (const int*)  d_in[1];   // [2,E]
    const int*   uni_ei = (const int*)  d_in[2];
    const float* bn_g   = (const float*)d_in[3];
    const float* bn_b   = (const float*)d_in[4];
    const float* enc_w1 = (const float*)d_in[5];
    const float* enc_b1 = (const float*)d_in[6];
    const float* enc_w2 = (const float*)d_in[7];
    const float* enc_b2 = (const float*)d_in[8];
    const float* gi_w   = (const float*)d_in[9];
    const float* gi_as  = (const float*)d_in[10];
    const float* gi_ad  = (const float*)d_in[11];
    const float* gi_b   = (const float*)d_in[12];
    const float* gu_w   = (const float*)d_in[13];
    const float* gu_as  = (const float*)d_in[14];
    const float* gu_ad  = (const float*)d_in[15];
    const float* gu_b   = (const float*)d_in[16];
    const float* dec_w1 = (const float*)d_in[17];
    const float* dec_b1 = (const float*)d_in[18];
    const float* dec_w2 = (const float*)d_in[19];
    const float* dec_b2 = (const float*)d_in[20];
    const float* fa_w   = (const float*)d_in[21];
    const float* fa_b   = (const float*)d_in[22];

    float* out = (float*)d_out;                 // [0,N): deep_factor, [N, N+N*F): attn

    // ---- workspace carve-up (floats) ----
    float* w = (float*)d_ws;
    float* psum  = w; w += NBN * FF;
    float* psq   = w; w += NBN * FF;
    float* scale = w; w += FF;
    float* shift = w; w += FF;
    float* h1    = w; w += (size_t)NN * FF;     // encoder hidden
    float* Cenc  = w; w += (size_t)NN * HID;
    float* C_I   = w; w += (size_t)NN * HID;
    float* C_U   = w; w += (size_t)NN * HID;
    float* xl    = w; w += (size_t)NN * HEADS * HID;   // GAT lin output (reused)
    float* a_src = w; w += (size_t)NN * HEADS;
    float* a_dst = w; w += (size_t)NN * HEADS;
    unsigned* emax = (unsigned*)w; w += (size_t)NN * HEADS;
    float* denom = w; w += (size_t)NN * HEADS;
    float* ebuf  = w; w += (size_t)(EE + NN) * HEADS;
    float* accum = w; w += (size_t)NN * HEADS * HID;
    float* h2    = w; w += (size_t)NN * HID;
    (void)ws_size; (void)n_in; (void)in_sizes; (void)out_size;

    const int RB  = (NN / 16 + 3) / 4;          // 782 row blocks (4 waves x 16 rows)
    const int NEL = EE + NN;                    // edges + self loops

    // 1) BatchNorm statistics (application fused into enc1 A-load)
    bn_partial  <<<NBN, FF, 0, stream>>>(x, psum, psq);
    bn_finalize <<<1,   FF, 0, stream>>>(psum, psq, bn_g, bn_b, scale, shift);

    // 2) encoder: relu(bn(x) @ enc_w1 + b1) -> relu(h1 @ enc_w2 + b2)
    wmma_gemm<1, 1><<<dim3(RB, 2), 128, 0, stream>>>(x, nullptr, nullptr, scale, shift,
                                                     enc_w1, enc_b1, h1, FF, 2 * HID);
    wmma_gemm<1, 0><<<dim3(RB, 1), 128, 0, stream>>>(h1, nullptr, nullptr, nullptr, nullptr,
                                                     enc_w2, enc_b2, Cenc, 2 * HID, HID);

    // 3) industry GAT: Cenc -> C_I
    {
        const int* srcs = ind_ei; const int* dsts = ind_ei + EE;
        wmma_gemm<0, 0><<<dim3(RB, 2), 128, 0, stream>>>(Cenc, nullptr, nullptr, nullptr, nullptr,
                                                         gi_w, nullptr, xl, HID, HEADS * HID);
        att_dots   <<<(NN * HEADS + 255) / 256, 256, 0, stream>>>(xl, gi_as, gi_ad, a_src, a_dst);
        gat_init   <<<(NN * HEADS * HID + 255) / 256, 256, 0, stream>>>(emax, denom, accum);
        edge_phase1<<<(NEL + 255) / 256, 256, 0, stream>>>(srcs, dsts, a_src, a_dst, ebuf, emax);
        edge_phase2<<<(NEL + 255) / 256, 256, 0, stream>>>(dsts, ebuf, emax, denom);
        edge_phase3<<<(NEL + 3) / 4, 128, 0, stream>>>(srcs, dsts, ebuf, denom, xl, accum);
        gat_finalize<<<(NN * HID + 255) / 256, 256, 0, stream>>>(accum, gi_b, Cenc, C_I);
    }

    // 4) universe GAT: C_I -> C_U
    {
        const int* srcs = uni_ei; const int* dsts = uni_ei + EE;
        wmma_gemm<0, 0><<<dim3(RB, 2), 128, 0, stream>>>(C_I, nullptr, nullptr, nullptr, nullptr,
                                                         gu_w, nullptr, xl, HID, HEADS * HID);
        att_dots   <<<(NN * HEADS + 255) / 256, 256, 0, stream>>>(xl, gu_as, gu_ad, a_src, a_dst);
        gat_init   <<<(NN * HEADS * HID + 255) / 256, 256, 0, stream>>>(emax, denom, accum);
        edge_phase1<<<(NEL + 255) / 256, 256, 0, stream>>>(srcs, dsts, a_src, a_dst, ebuf, emax);
        edge_phase2<<<(NEL + 255) / 256, 256, 0, stream>>>(dsts, ebuf, emax, denom);
        edge_phase3<<<(NEL + 3) / 4, 128, 0, stream>>>(srcs, dsts, ebuf, denom, xl, accum);
        gat_finalize<<<(NN * HID + 255) / 256, 256, 0, stream>>>(accum, gu_b, C_I, C_U);
    }

    // 5) decoder: leaky([Cenc|C_I|C_U] @ dec_w1 + b1) fused concat -> dot with dec_w2
    wmma_gemm<2, 2><<<dim3(RB, 1), 128, 0, stream>>>(Cenc, C_I, C_U, nullptr, nullptr,
                                                     dec_w1, dec_b1, h2, 3 * HID, HID);
    dec2<<<(NN + 255) / 256, 256, 0, stream>>>(h2, dec_w2, dec_b2, out);

    // 6) factor attention: leaky(x @ fa_w + fa_b) -> row softmax, in-place in d_out
    float* attn = out + NN;
    wmma_gemm<2, 0><<<dim3(RB, 2), 128, 0, stream>>>(x, nullptr, nullptr, nullptr, nullptr,
                                                     fa_w, fa_b, attn, FF, FF);
    softmax_rows<<<NN, FF, 0, stream>>>(attn);
}